// PCENTransform_24249385353277
// MI455X (gfx1250) — compile-verified
//
#include <hip/hip_runtime.h>
#include <hip/hip_bf16.h>

typedef float v2f __attribute__((ext_vector_type(2)));
typedef float v8f __attribute__((ext_vector_type(8)));

// Problem constants (fixed by the reference)
constexpr int   Bn    = 32;
constexpr int   Tn    = 8192;
constexpr int   Fn    = 128;
constexpr float S_    = 0.025f;
constexpr float A_    = 1.0f - S_;       // 0.975
constexpr float ALPHA = 0.98f;
constexpr float DELTA = 2.0f;
constexpr float EPSI  = 1e-6f;
constexpr float SQRT_DELTA = 1.41421356237309515f; // sqrt(2)

// Chunking: warm-up of 512 steps -> carried-state error a^512 ~ 2.4e-6 (below f32 tolerance)
constexpr int CHUNK  = 1024;
constexpr int WARM   = 512;
constexpr int NCHUNK = Tn / CHUNK;       // 8

__global__ __launch_bounds__(256) void pcen_wmma_kernel(const float* __restrict__ x,
                                                        float* __restrict__ out) {
    const int lane  = threadIdx.x & 31;
    const int wave  = threadIdx.x >> 5;       // 8 waves -> 8 f-tiles (F=128)
    const int h     = lane >> 4;              // half-wave select
    const int fm    = lane & 15;              // column f within tile == row m for A operand
    const int b     = blockIdx.x / NCHUNK;
    const int chunk = blockIdx.x % NCHUNK;
    const int fbase = wave * 16;

    // powers of a (need up to a^16 for the carry column)
    float apow[18];
    apow[0] = 1.0f;
#pragma unroll
    for (int i = 1; i < 18; ++i) apow[i] = apow[i - 1] * A_;

    // A-operands in ISA 16x4 layout: V0 <-> K = 4k+2h, V1 <-> K = 4k+2h+1, row m = lane&15
    v2f La[4];  // lower-triangular EMA matrix L[i][j] = s * a^(i-j)  (i >= j)
    v2f Ia[4];  // identity (for B-layout -> D-layout relayout of X)
#pragma unroll
    for (int k = 0; k < 4; ++k) {
        const int j0 = 4 * k + 2 * h;
        const int j1 = j0 + 1;
        La[k].x = (fm >= j0) ? S_ * apow[fm - j0] : 0.0f;
        La[k].y = (fm >= j1) ? S_ * apow[fm - j1] : 0.0f;
        Ia[k].x = (fm == j0) ? 1.0f : 0.0f;
        Ia[k].y = (fm == j1) ? 1.0f : 0.0f;
    }

    // Carry column p[i] = a^(row+1), row = i + 8h (C/D layout)
    float pA[8];
#pragma unroll
    for (int i = 0; i < 8; ++i) pA[i] = apow[i + 8 * h + 1];

    const int t_out0  = chunk * CHUNK;
    const int t_warm0 = (chunk == 0) ? 0 : (t_out0 - WARM);
    const int t_end   = t_out0 + CHUNK;

    const float* xb_base = x   + (size_t)b * Tn * Fn + fbase;
    float*       ob_base = out + (size_t)b * Tn * Fn + fbase;

    // Exact series start: M[0] = x[0] is reproduced by m_prev = x[0] since a+s = 1.
    // Warm-up chunks start from 0 (error decays by a^WARM).
    float carry_m = (chunk == 0) ? xb_base[fm] : 0.0f;

    const int r0 = 2 * h;  // B-layout row offset within each K-group of 4

    for (int t0 = t_warm0; t0 < t_end; t0 += 16) {
        // Stream-ahead hint (global_prefetch_b8), ~3 tiles ahead
        __builtin_prefetch(xb_base + (size_t)(t0 + 48) * Fn + fm, 0, 0);

        // Load X tile (16t x 16f) directly in WMMA B layout:
        // xb[2k] = X[4k+2h][f], xb[2k+1] = X[4k+2h+1][f]
        float xb[8];
#pragma unroll
        for (int k = 0; k < 4; ++k) {
            const float* p = xb_base + (size_t)(t0 + 4 * k + r0) * Fn + fm;
            xb[2 * k]     = p[0];
            xb[2 * k + 1] = p[Fn];
        }

        // acc init = p (x) m_prev  (rank-1 carry term), then M = L * X + acc
        const float m_f = carry_m;
        v8f acc;
#pragma unroll
        for (int i = 0; i < 8; ++i) acc[i] = pA[i] * m_f;

#pragma unroll
        for (int k = 0; k < 4; ++k) {
            v2f bk; bk.x = xb[2 * k]; bk.y = xb[2 * k + 1];
            acc = __builtin_amdgcn_wmma_f32_16x16x4_f32(
                false, La[k], false, bk, (short)0, acc, false, false);
        }

        if (t0 >= t_out0) {
            // Relayout X from B layout to C/D layout: X_D = I * X_B
            v8f xd = {};
#pragma unroll
            for (int k = 0; k < 4; ++k) {
                v2f bk; bk.x = xb[2 * k]; bk.y = xb[2 * k + 1];
                xd = __builtin_amdgcn_wmma_f32_16x16x4_f32(
                    false, Ia[k], false, bk, (short)0, xd, false, false);
            }
            // PCEN pointwise: (x * (M+eps)^-alpha + delta)^0.5 - delta^0.5
            float* op = ob_base + (size_t)(t0 + 8 * h) * Fn + fm;
#pragma unroll
            for (int i = 0; i < 8; ++i) {
                const float Mv   = acc[i] + EPSI;
                // (M+eps)^(-alpha) = exp2(-alpha * log2(M+eps)) : v_log_f32 + v_exp_f32
                const float gain = __builtin_amdgcn_exp2f(-ALPHA * __builtin_amdgcn_logf(Mv));
                const float y    = __builtin_amdgcn_sqrtf(xd[i] * gain + DELTA) - SQRT_DELTA;
                op[(size_t)i * Fn] = y;
            }
        }

        // Carry = M[row 15] -> lives in VGPR7, lanes 16..31; broadcast to all lanes of column f
        carry_m = __shfl(acc[7], fm | 16, 32);
    }
}

extern "C" void kernel_launch(void* const* d_in, const int* in_sizes, int n_in,
                              void* d_out, int out_size, void* d_ws, size_t ws_size,
                              hipStream_t stream) {
    (void)in_sizes; (void)n_in; (void)d_ws; (void)ws_size; (void)out_size;
    const float* xin = (const float*)d_in[0];
    float* yout = (float*)d_out;
    dim3 grid(Bn * NCHUNK);  // 256 blocks
    dim3 block(256);         // 8 waves, one 16-wide f-tile each
    pcen_wmma_kernel<<<grid, block, 0, stream>>>(xin, yout);
}